// SAGE_13159779795716
// MI455X (gfx1250) — compile-verified
//
#include <hip/hip_runtime.h>
#include <hip/hip_bf16.h>

typedef float v2f __attribute__((ext_vector_type(2)));
typedef float v8f __attribute__((ext_vector_type(8)));
typedef unsigned int u32x4 __attribute__((ext_vector_type(4)));
typedef int i32x4 __attribute__((ext_vector_type(4)));
typedef int i32x8 __attribute__((ext_vector_type(8)));

#define NNODES 50000
#define HDIM   128
#define NLAB   64
#define BN_EPS 1e-5f
#define ROWT   4        // row tiles (of 16 rows) per block

#if defined(__AMDGCN__) && __has_builtin(__builtin_amdgcn_tensor_load_to_lds) && \
    __has_builtin(__builtin_amdgcn_s_wait_tensorcnt)
#define USE_TDM 1
#else
#define USE_TDM 0
#endif

#if USE_TDM
// ---------------------------------------------------------------------------
// Issue one TDM descriptor: load a 16x128 f32 tile (row stride 128 elements)
// into LDS, with 4 dwords of LDS padding every 128 dwords so the DMA directly
// produces the bank-conflict-free As[16][132] layout.  CDNA5 ISA.
// ---------------------------------------------------------------------------
__device__ __forceinline__ void tdm_load_tile(const float* gsrc, void* lds_dst) {
  const unsigned lds = (unsigned)(size_t)lds_dst;            // LDS offset = addr[31:0]
  const unsigned long long ga = (unsigned long long)(size_t)gsrc;

  u32x4 g0;
  g0[0] = 1u;                                               // count=1, user mode
  g0[1] = lds;                                              // lds_addr
  g0[2] = (unsigned)ga;                                     // global_addr[31:0]
  g0[3] = (unsigned)((ga >> 32) & 0x1FFFFFFull)             // global_addr[56:32]
        | (2u << 30);                                       // type=2 ("image")

  // group1 bit packing (256 bits as 4x u64)
  const unsigned long long q0 =
        (2ull << 16)                                        // data_size = 4B
      | (1ull << 20)                                        // pad_enable
      | (6ull << 22)                                        // pad_interval = 128 dwords
      | (3ull << 25)                                        // pad_amount  = 4 dwords
      | ((128ull & 0xFFFFull) << 48);                       // tensor_dim0[15:0] = 128
  const unsigned long long q1 =
        (128ull >> 16)                                      // tensor_dim0[31:16]
      | (65536ull << 16)                                    // tensor_dim1 (large, no OOB)
      | (128ull << 48);                                     // tile_dim0 = 128
  const unsigned long long q2 =
        16ull                                               // tile_dim1 = 16 rows
      | (128ull << 32);                                     // tensor_dim0_stride[31:0]=128
  const unsigned long long q3 = 0ull;                       // stride hi / dim1_stride = 0

  i32x8 g1;
  g1[0] = (int)q0; g1[1] = (int)(q0 >> 32);
  g1[2] = (int)q1; g1[3] = (int)(q1 >> 32);
  g1[4] = (int)q2; g1[5] = (int)(q2 >> 32);
  g1[6] = (int)q3; g1[7] = (int)(q3 >> 32);

  i32x4 gz = {0, 0, 0, 0};
#if __clang_major__ >= 23
  i32x8 gz8 = {0, 0, 0, 0, 0, 0, 0, 0};
  __builtin_amdgcn_tensor_load_to_lds(g0, g1, gz, gz, gz8, 0);
#else
  __builtin_amdgcn_tensor_load_to_lds(g0, g1, gz, gz, 0);
#endif
}
#endif  // USE_TDM

// ---------------------------------------------------------------------------
// GEMM: C[N,Hout] = A[N,128] @ W[128,Hout] + biasA + biasB
// Block handles ROWT row-tiles of 16 rows; one wave per 16-col tile.
// B fragments (64 VGPRs/wave) preloaded once, reused across row tiles.
// A tiles double-buffered in LDS, staged by the Tensor Data Mover.
// Inner loop is pure ds_load_b64 + v_wmma_f32_16x16x4_f32.
// ---------------------------------------------------------------------------
__global__ void sage_gemm_wmma(const float* __restrict__ A,
                               const float* __restrict__ W,
                               const float* __restrict__ biasA,
                               const float* __restrict__ biasB,
                               float* __restrict__ C,
                               int Hout, int nRowTiles) {
  __shared__ float As[2][16][132];                 // +4 pad per row
  const int lane = threadIdx.x & 31;
  const int wave = threadIdx.x >> 5;
  const int col0 = wave << 4;
  const int m    = lane & 15;
  const int kh   = lane >> 4;

  const int t0   = blockIdx.x * ROWT;
  const int tEnd = (t0 + ROWT < nRowTiles) ? (t0 + ROWT) : nRowTiles;

  // Preload this wave's full B operand into registers (32 k-steps x v2f).
  v2f bfrag[32];
#pragma unroll
  for (int ks = 0; ks < 32; ++ks) {
    const int kk = 4 * ks + 2 * kh;
    bfrag[ks].x = W[kk * Hout + col0 + m];
    bfrag[ks].y = W[(kk + 1) * Hout + col0 + m];
  }

  const int col = col0 + m;
  float bsum = 0.0f;
  if (biasA) bsum += biasA[col];
  if (biasB) bsum += biasB[col];

#if USE_TDM
  if (threadIdx.x == 0)
    tdm_load_tile(A + (size_t)t0 * 16 * 128, &As[t0 & 1][0][0]);
#endif

  for (int t = t0; t < tEnd; ++t) {
    const int buf = t & 1;
#if USE_TDM
    const bool more = (t + 1) < tEnd;
    if (more && threadIdx.x == 0)
      tdm_load_tile(A + (size_t)(t + 1) * 16 * 128, &As[buf ^ 1][0][0]);
    if (more) __builtin_amdgcn_s_wait_tensorcnt(1);   // oldest (tile t) done
    else      __builtin_amdgcn_s_wait_tensorcnt(0);
    __syncthreads();                                  // release all waves
#else
    __syncthreads();                                  // prev reads done
    for (int i = threadIdx.x; i < 16 * 128; i += blockDim.x)
      As[buf][i >> 7][i & 127] = A[((size_t)t * 16 + (i >> 7)) * 128 + (i & 127)];
    __syncthreads();
#endif

    v8f acc = {};
#pragma unroll
    for (int ks = 0; ks < 32; ++ks) {
      const int kk = 4 * ks + 2 * kh;
      const float2 av = *(const float2*)&As[buf][m][kk];     // ds_load_b64
      v2f a; a.x = av.x; a.y = av.y;
      acc = __builtin_amdgcn_wmma_f32_16x16x4_f32(
          false, a, false, bfrag[ks], (short)0, acc, false, false);
    }

    // C/D layout: VGPR r -> row (16t + r + 8*kh), col = col0 + m.
    const int rowb = t * 16 + kh * 8;
#pragma unroll
    for (int r = 0; r < 8; ++r)
      C[(size_t)(rowb + r) * Hout + col] = acc[r] + bsum;

#if USE_TDM
    __syncthreads();   // reads of As[buf] done before next iter's prefetch
#endif
  }
}

// ---------------------------------------------------------------------------
// SPMM scatter: acc[dst] += w * msg[src]; H/4 lanes per edge, float4 gather,
// 4x global_atomic_add_f32 scatter (accumulators are L2-resident).
// ---------------------------------------------------------------------------
__global__ void sage_spmm_scatter(const float* __restrict__ msg,
                                  const int* __restrict__ src,
                                  const int* __restrict__ dst,
                                  const float* __restrict__ ew,
                                  float* __restrict__ acc,
                                  int E, int H, int shift) {
  const long long tid = (long long)blockIdx.x * blockDim.x + threadIdx.x;
  const int e = (int)(tid >> shift);
  if (e >= E) return;
  const int f = ((int)tid & ((1 << shift) - 1)) << 2;
  const float wt = ew[e];
  const float4 v = *(const float4*)(msg + (long long)src[e] * H + f);
  float* p = acc + (long long)dst[e] * H + f;
  atomicAdd(p + 0, v.x * wt);
  atomicAdd(p + 1, v.y * wt);
  atomicAdd(p + 2, v.z * wt);
  atomicAdd(p + 3, v.w * wt);
}

// ---------------------------------------------------------------------------
// BatchNorm column stats over N rows (H = 128 fixed).
// ---------------------------------------------------------------------------
__global__ void sage_bn_stats(const float* __restrict__ h,
                              float* __restrict__ sums,
                              float* __restrict__ sumsq, int N) {
  __shared__ float s1[256], s2[256];
  const int col  = threadIdx.x & 127;
  const int half = threadIdx.x >> 7;
  float a = 0.0f, b = 0.0f;
  for (int row = blockIdx.x * 2 + half; row < N; row += gridDim.x * 2) {
    const float v = h[row * 128 + col];
    a += v; b += v * v;
  }
  s1[threadIdx.x] = a; s2[threadIdx.x] = b;
  __syncthreads();
  if (threadIdx.x < 128) {
    atomicAdd(&sums[col],  s1[threadIdx.x] + s1[threadIdx.x + 128]);
    atomicAdd(&sumsq[col], s2[threadIdx.x] + s2[threadIdx.x + 128]);
  }
}

__global__ void sage_bn_finalize(const float* __restrict__ sums,
                                 const float* __restrict__ sumsq,
                                 const float* __restrict__ gamma,
                                 const float* __restrict__ beta,
                                 float* __restrict__ scale,
                                 float* __restrict__ shift, float invN) {
  const int c = threadIdx.x;
  const float mean = sums[c] * invN;
  const float var  = sumsq[c] * invN - mean * mean;
  const float a    = gamma[c] * __frsqrt_rn(var + BN_EPS);
  scale[c] = a;
  shift[c] = beta[c] - mean * a;
}

__device__ __forceinline__ unsigned sage_hash(unsigned x) {
  x = (x ^ 61u) ^ (x >> 16);
  x *= 9u;  x ^= x >> 4;
  x *= 0x27d4eb2du;  x ^= x >> 15;
  return x;
}

// BN apply + ReLU + dropout(0.5, 1/(1-p) scaling), float4 vectorized, in place.
__global__ void sage_bn_relu_drop(float* __restrict__ h,
                                  const float* __restrict__ scale,
                                  const float* __restrict__ shift,
                                  int total4, unsigned seed) {
  const int i = blockIdx.x * blockDim.x + threadIdx.x;
  if (i >= total4) return;
  float4 v = ((const float4*)h)[i];
  const int c0 = (i << 2) & 127;           // 4 consecutive cols, no wrap
  float o[4] = {v.x, v.y, v.z, v.w};
#pragma unroll
  for (int j = 0; j < 4; ++j) {
    float t = fmaxf(o[j] * scale[c0 + j] + shift[c0 + j], 0.0f);
    const unsigned r = sage_hash((unsigned)(i * 4 + j) ^ seed);
    o[j] = (r & 0x10000u) ? t * 2.0f : 0.0f;
  }
  ((float4*)h)[i] = make_float4(o[0], o[1], o[2], o[3]);
}

// out[i, :] = acc[idx[i], :] + bias   (NL = 64), float4 vectorized.
__global__ void sage_gather_out(const float* __restrict__ acc,
                                const int* __restrict__ idx,
                                const float* __restrict__ bias,
                                float* __restrict__ out, int total4) {
  const int t = blockIdx.x * blockDim.x + threadIdx.x;
  if (t >= total4) return;
  const int i = t >> 4, c = (t & 15) << 2;
  const float4 v = *(const float4*)(acc + (long long)idx[i] * NLAB + c);
  const float4 b = *(const float4*)(bias + c);
  ((float4*)out)[t] = make_float4(v.x + b.x, v.y + b.y, v.z + b.z, v.w + b.w);
}

// ---------------------------------------------------------------------------
extern "C" void kernel_launch(void* const* d_in, const int* in_sizes, int n_in,
                              void* d_out, int out_size, void* d_ws, size_t ws_size,
                              hipStream_t stream) {
  const float* x        = (const float*)d_in[0];
  const int*   esrc     = (const int*)  d_in[1];
  const int*   edst     = (const int*)  d_in[2];
  const float* ew       = (const float*)d_in[3];
  const int*   idx      = (const int*)  d_in[4];
  const float* W_node0  = (const float*)d_in[5];
  const float* b_node0  = (const float*)d_in[6];
  const float* W_neigh0 = (const float*)d_in[7];
  const float* b_neigh0 = (const float*)d_in[8];
  const float* bias0    = (const float*)d_in[9];
  const float* gamma0   = (const float*)d_in[10];
  const float* beta0    = (const float*)d_in[11];
  const float* W_node1  = (const float*)d_in[12];
  const float* W_neigh1 = (const float*)d_in[13];
  const float* bias1    = (const float*)d_in[14];
  const float* gamma1   = (const float*)d_in[15];
  const float* beta1    = (const float*)d_in[16];
  const float* W_node2  = (const float*)d_in[17];
  const float* W_neigh2 = (const float*)d_in[18];
  const float* bias2    = (const float*)d_in[19];
  float* out = (float*)d_out;

  const int E    = in_sizes[1];
  const int NIDX = in_sizes[4];
  const int N    = NNODES;

  char* ws = (char*)d_ws;
  size_t off = 0;
  auto carve = [&](size_t bytes) {
    void* p = ws + off;
    off += (bytes + 255) & ~(size_t)255;
    return p;
  };
  float* bufA = (float*)carve(sizeof(float) * (size_t)N * HDIM);
  float* bufB = (float*)carve(sizeof(float) * (size_t)N * HDIM);
  float* bufC = (float*)carve(sizeof(float) * (size_t)N * HDIM);
  float* bufD = (float*)carve(sizeof(float) * (size_t)N * NLAB);
  float* sums  = (float*)carve(sizeof(float) * HDIM);
  float* sumsq = (float*)carve(sizeof(float) * HDIM);
  float* scl   = (float*)carve(sizeof(float) * HDIM);
  float* shf   = (float*)carve(sizeof(float) * HDIM);

  const int rowTiles    = N / 16;                       // 3125
  const int gemm_blocks = (rowTiles + ROWT - 1) / ROWT; // 782
  const long long spmm128 = (long long)E * 32;
  const long long spmm64  = (long long)E * 16;

  // ---- layer 0 ----
  sage_gemm_wmma<<<gemm_blocks, 256, 0, stream>>>(x, W_node0,  b_node0,  bias0,  bufA, 128, rowTiles);
  sage_gemm_wmma<<<gemm_blocks, 256, 0, stream>>>(x, W_neigh0, b_neigh0, nullptr, bufB, 128, rowTiles);
  sage_spmm_scatter<<<(int)((spmm128 + 255) / 256), 256, 0, stream>>>(
      bufB, esrc, edst, ew, bufA, E, 128, 5);
  hipMemsetAsync(sums,  0, sizeof(float) * HDIM, stream);
  hipMemsetAsync(sumsq, 0, sizeof(float) * HDIM, stream);
  sage_bn_stats<<<512, 256, 0, stream>>>(bufA, sums, sumsq, N);
  sage_bn_finalize<<<1, 128, 0, stream>>>(sums, sumsq, gamma0, beta0, scl, shf, 1.0f / N);
  sage_bn_relu_drop<<<(N * 32 + 255) / 256, 256, 0, stream>>>(
      bufA, scl, shf, N * 32, 0x13579BDFu);

  // ---- layer 1 ----
  sage_gemm_wmma<<<gemm_blocks, 256, 0, stream>>>(bufA, W_node1,  bias1,  nullptr, bufC, 128, rowTiles);
  sage_gemm_wmma<<<gemm_blocks, 256, 0, stream>>>(bufA, W_neigh1, nullptr, nullptr, bufB, 128, rowTiles);
  sage_spmm_scatter<<<(int)((spmm128 + 255) / 256), 256, 0, stream>>>(
      bufB, esrc, edst, ew, bufC, E, 128, 5);
  hipMemsetAsync(sums,  0, sizeof(float) * HDIM, stream);
  hipMemsetAsync(sumsq, 0, sizeof(float) * HDIM, stream);
  sage_bn_stats<<<512, 256, 0, stream>>>(bufC, sums, sumsq, N);
  sage_bn_finalize<<<1, 128, 0, stream>>>(sums, sumsq, gamma1, beta1, scl, shf, 1.0f / N);
  sage_bn_relu_drop<<<(N * 32 + 255) / 256, 256, 0, stream>>>(
      bufC, scl, shf, N * 32, 0x2468ACE1u);

  // ---- conv_last: project to NL=64, bias2 folded into the gather ----
  sage_gemm_wmma<<<gemm_blocks, 128, 0, stream>>>(bufC, W_node2,  nullptr, nullptr, bufD, 64, rowTiles);
  sage_gemm_wmma<<<gemm_blocks, 128, 0, stream>>>(bufC, W_neigh2, nullptr, nullptr, bufA, 64, rowTiles);
  sage_spmm_scatter<<<(int)((spmm64 + 255) / 256), 256, 0, stream>>>(
      bufA, esrc, edst, ew, bufD, E, 64, 4);
  sage_gather_out<<<(NIDX * 16 + 255) / 256, 256, 0, stream>>>(
      bufD, idx, bias2, out, NIDX * 16);
}